// CharEmbedding_6433861009592
// MI455X (gfx1250) — compile-verified
//
#include <hip/hip_runtime.h>

typedef __attribute__((ext_vector_type(2))) float v2f;
typedef __attribute__((ext_vector_type(8))) float v8f;

#define VOCAB   262
#define EMBD    64
#define OUT_CHS 100
#define BS      32
#define SEQ     512
#define WORD    16

#define VROWS   60     // w' rows
#define VCOLS   112    // c' columns padded to 7 full 16-wide WMMA tiles

// ---------------------------------------------------------------------------
// Kernel 1: embd[b,s,d] = sum_{i<16} emb_table[x[b,s,i], d]
// One 64-thread block per (b,s); x indices are wave-uniform -> scalar loads.
// ---------------------------------------------------------------------------
__global__ __launch_bounds__(64) void embed_sum_kernel(
    const int* __restrict__ x, const float* __restrict__ table,
    float* __restrict__ embd)
{
    const int bs = blockIdx.x;           // b*512 + s
    const int d  = threadIdx.x;          // 0..63
    const int* xw = x + bs * WORD;
    float acc = 0.f;
    #pragma unroll
    for (int i = 0; i < WORD; ++i) {
        acc += table[xw[i] * EMBD + d];
    }
    embd[(size_t)bs * EMBD + d] = acc;
}

// ---------------------------------------------------------------------------
// Kernel 2: one block per (b, h').
//   The 6000 flat y-elements feeding out[b,h',:] are the 100 contiguous conv
//   rows rr = h'*100 .. h'*100+99  (c = rr>>9, h = rr&511), width 60 each.
//   Stage the 100 needed embd rows in LDS, compute relu(conv) into a padded
//   60x112 LDS matrix (row w', col c'; cols 100..111 zero), then column-sum
//   with V_WMMA_F32_16X16X4_F32 using an all-ones A. Padding makes every
//   B-operand DS load unconditional (no EXEC predication in the WMMA chain).
// ---------------------------------------------------------------------------
__global__ __launch_bounds__(128) void conv_scramble_sum_kernel(
    const float* __restrict__ embd, const float* __restrict__ conv_w,
    const float* __restrict__ conv_b, float* __restrict__ out)
{
    __shared__ float sE[100 * 64];         // 25.6 KB: 100 embd rows
    __shared__ float sV[VROWS * VCOLS];    // 26.9 KB: padded 60x112 value matrix

    const int hp  = blockIdx.x;      // h' in 0..511
    const int b   = blockIdx.y;      // 0..31
    const int tid = threadIdx.x;     // 0..127 (4 waves)
    const int rr0 = hp * 100;

    // Stage the 100 embd rows this block needs (h = (rr0+j) & 511).
    const float* ebase = embd + (size_t)b * SEQ * EMBD;
    for (int idx = tid; idx < 100 * 64; idx += 128) {
        const int j = idx >> 6;
        const int d = idx & 63;
        const int h = (rr0 + j) & 511;
        sE[idx] = ebase[h * 64 + d];
    }

    // Zero the 12 padding columns (60 rows x 12 cols = 720 elements).
    for (int q = tid; q < VROWS * (VCOLS - OUT_CHS); q += 128) {
        const int row = q / (VCOLS - OUT_CHS);
        const int cp  = q - row * (VCOLS - OUT_CHS);
        sV[row * VCOLS + OUT_CHS + cp] = 0.f;
    }

    __syncthreads();

    // relu(conv) at flat offset p = j*60 + w; scatter to padded (w',c') slot:
    //   w' = p/100, c' = p%100  ->  sV[w'*112 + c'].
    for (int p = tid; p < 6000; p += 128) {
        const int j  = p / 60;
        const int w  = p - j * 60;
        const int rr = rr0 + j;
        const int c  = rr >> 9;                   // conv channel (1-2 per block)
        const float* wt = conv_w + c * 5;
        float acc = conv_b[c];
        #pragma unroll
        for (int k = 0; k < 5; ++k)
            acc = fmaf(sE[j * 64 + w + k], wt[k], acc);
        const int wp = p / 100;
        const int cp = p - wp * 100;
        sV[wp * VCOLS + cp] = fmaxf(acc, 0.f);
    }
    __syncthreads();

    // Column-sum of sV (60 rows x 112 cols) via f32 WMMA:
    //   D = ones(16x4) * B(4x16) accumulated over 15 K-chunks of 4 rows.
    // A 16x4 f32 = 2 VGPRs (K=0/2 in v0 halves, K=1/3 in v1); B 4x16 mirrors.
    // Wave-uniform control flow keeps EXEC all-ones through the WMMA chain.
    const int wave = tid >> 5;
    const int lane = tid & 31;
    const int half = lane >> 4;
    const int n    = lane & 15;

    const v2f aones = {1.f, 1.f};

    for (int nt = wave; nt < 7; nt += 4) {        // 7 tiles = 112 cols
        const int col = nt * 16 + n;
        v8f acc = {};
        #pragma unroll
        for (int kb = 0; kb < 15; ++kb) {
            const int r = kb * 4 + half * 2;      // rows r, r+1 for this lane half
            v2f bvec;
            bvec.x = sV[r * VCOLS + col];
            bvec.y = sV[(r + 1) * VCOLS + col];
            acc = __builtin_amdgcn_wmma_f32_16x16x4_f32(
                false, aones, false, bvec, (short)0, acc, false, false);
        }
        // D row M=0 lives in acc[0] on lanes 0..15 (N = lane).
        if (lane < 16 && col < OUT_CHS)
            out[((size_t)b * SEQ + hp) * OUT_CHS + col] = acc[0];
    }
}

// ---------------------------------------------------------------------------
extern "C" void kernel_launch(void* const* d_in, const int* in_sizes, int n_in,
                              void* d_out, int out_size, void* d_ws, size_t ws_size,
                              hipStream_t stream) {
    const int*   x      = (const int*)d_in[0];    // (32,512,16) int32
    const float* table  = (const float*)d_in[1];  // (262,64)    f32
    const float* conv_w = (const float*)d_in[2];  // (100,1,1,5) f32
    const float* conv_b = (const float*)d_in[3];  // (100,)      f32
    float*       out    = (float*)d_out;          // (32,512,100) f32
    float*       embd   = (float*)d_ws;           // (32,512,64)  f32 = 4 MB scratch

    embed_sum_kernel<<<BS * SEQ, 64, 0, stream>>>(x, table, embd);

    dim3 grid(SEQ, BS);
    conv_scramble_sum_kernel<<<grid, 128, 0, stream>>>(embd, conv_w, conv_b, out);

    (void)in_sizes; (void)n_in; (void)out_size; (void)ws_size;
}